// CPCLoss_2748779070060
// MI455X (gfx1250) — compile-verified
//
#include <hip/hip_runtime.h>

typedef __attribute__((ext_vector_type(2))) float v2f;
typedef __attribute__((ext_vector_type(8))) float v8f;

#define B_   16
#define T_   1024
#define D_   128
#define K_   12
#define TK_  (T_ - K_)          // 1012
#define NNEG_ 128
#define TAU_ 0.07f
#define EPS_ 1e-8f
#define NPAIR_ (TK_ * B_)       // 16192
#define ROWS_ (B_ * T_)         // 16384

// ---------------------------------------------------------------------------
// Kernel 1: inverse norms for all embedding rows and all context rows.
// One wave (32 lanes) per row; each lane loads 4 floats, shfl-reduce sum-sq.
// ---------------------------------------------------------------------------
__global__ void __launch_bounds__(256)
cpc_norms_kernel(const float* __restrict__ ctx, const float* __restrict__ emb,
                 float* __restrict__ inv_c, float* __restrict__ inv_z) {
    int wave = (blockIdx.x * blockDim.x + threadIdx.x) >> 5;
    int lane = threadIdx.x & 31;
    if (wave >= 2 * ROWS_) return;                 // wave-uniform guard
    bool is_emb = (wave < ROWS_);
    int row = is_emb ? wave : (wave - ROWS_);
    const float* src = is_emb ? emb : ctx;
    const float4* p = (const float4*)(src + (size_t)row * D_);
    float4 v = p[lane];                            // 32 lanes * 4 floats = 128
    float s = v.x * v.x + v.y * v.y + v.z * v.z + v.w * v.w;
    #pragma unroll
    for (int m = 16; m; m >>= 1) s += __shfl_xor(s, m, 32);
    if (lane == 0) {
        float inv = 1.0f / fmaxf(sqrtf(s), EPS_);
        if (is_emb) inv_z[row] = inv; else inv_c[row] = inv;
    }
}

// ---------------------------------------------------------------------------
// Kernel 2: one wave per (t, b) pair.
//   - context row staged in LDS; WMMA B operand (replicated over all 16
//     N-columns) streams back via ds_load_2addr_b64
//   - 8 m-tiles of 16 gathered negatives via v_wmma_f32_16x16x4_f32
//   - ALL index/inv_z loads and prefetches hoisted before the main loop so
//     the steady state has no dependent scalar loads: each iteration issues
//     the next tile's 32-deep global_load_b64 clause, then computes the
//     current tile, letting the compiler retire data with partial
//     s_wait_loadcnt while the next clause stays in flight.
//   - logits = D * inv_c / tau ; direct sum of exp (|logit| <= 1/tau = 14.3,
//     so the f32 exp-sum cannot overflow; max-subtraction unnecessary)
// ---------------------------------------------------------------------------
__global__ void __launch_bounds__(256)
cpc_pair_kernel(const float* __restrict__ ctx, const float* __restrict__ emb,
                const int* __restrict__ negidx,
                const float* __restrict__ inv_c, const float* __restrict__ inv_z,
                float* __restrict__ pair_loss) {
    __shared__ float c_lds[8 * D_];                // 8 waves * 512 B
    int lane = threadIdx.x & 31;
    int wid  = threadIdx.x >> 5;
    int p = blockIdx.x * 8 + wid;                  // pair id = t*B + b
    if (p >= NPAIR_) return;                       // wave-uniform
    int t = p / B_;
    int b = p - t * B_;

    const float* crow = ctx + ((size_t)b * T_ + t) * D_;
    const float* zpos = emb + ((size_t)b * T_ + (t + K_)) * D_;

    // Stage context row to LDS; compute positive dot in the same pass.
    float4 c4 = ((const float4*)crow)[lane];
    float4 z4 = ((const float4*)zpos)[lane];
    ((float4*)(c_lds + wid * D_))[lane] = c4;
    float pos = c4.x * z4.x + c4.y * z4.y + c4.z * z4.z + c4.w * z4.w;
    #pragma unroll
    for (int m = 16; m; m >>= 1) pos += __shfl_xor(pos, m, 32);
    __syncthreads();

    float ic  = inv_c[b * T_ + t];
    float izp = inv_z[b * T_ + (t + K_)];
    float pos_logit = pos * ic * izp * (1.0f / TAU_);
    float scale_out = ic * (1.0f / TAU_);

    const int* nidx = negidx + ((size_t)t * B_ + b) * NNEG_;
    const float* cl = c_lds + wid * D_;
    int koff = (lane >> 4) << 1;                   // 0 (lanes 0-15) or 2 (16-31)
    int mrow = lane & 15;                          // A-matrix row for this lane

    // ---- hoist ALL index-dependent loads out of the steady-state loop ----
    int   idxs[8];
    float izs[8];
    #pragma unroll
    for (int mt = 0; mt < 8; ++mt) {
        int idx = nidx[mt * 16 + mrow];
        idxs[mt] = min(max(idx, 0), ROWS_ - 1);
    }
    #pragma unroll
    for (int mt = 0; mt < 8; ++mt) izs[mt] = inv_z[idxs[mt]];
    #pragma unroll
    for (int mt = 1; mt < 8; ++mt)                 // warm caches well ahead
        __builtin_prefetch(emb + (size_t)idxs[mt] * D_, 0, 3);

    // Double-buffered A fragments: buf[i][k] = {z[4k+koff], z[4k+koff+1]}
    v2f buf[2][32];
    {
        const float* zr = emb + (size_t)idxs[0] * D_ + koff;
        #pragma unroll
        for (int j = 0; j < 32; ++j) buf[0][j] = *(const v2f*)(zr + j * 4);
    }

    float ssum = 0.0f;
    #pragma unroll
    for (int mt = 0; mt < 8; ++mt) {
        int cur = mt & 1;
        int nxt = cur ^ 1;
        // issue next tile's 32-deep gather clause before computing this tile
        if (mt < 7) {
            const float* zrn = emb + (size_t)idxs[mt + 1] * D_ + koff;
            #pragma unroll
            for (int j = 0; j < 32; ++j) buf[nxt][j] = *(const v2f*)(zrn + j * 4);
        }

        float izc = izs[mt];
        v8f acc = {};
        #pragma unroll
        for (int k = 0; k < 32; ++k) {
            v2f a = buf[cur][k];
            a.x *= izc;                            // fold in 1/||z||
            a.y *= izc;
            v2f bb = *(const v2f*)(cl + k * 4 + koff); // B: context, all cols equal
            acc = __builtin_amdgcn_wmma_f32_16x16x4_f32(
                false, a, false, bb, (short)0, acc, false, false);
        }
        // D layout: VGPR v holds rows v (lanes 0-15) / v+8 (lanes 16-31);
        // every tile logit replicated across 16 lanes (divided out after reduce)
        #pragma unroll
        for (int v = 0; v < 8; ++v) {
            ssum += __expf(acc[v] * scale_out);
        }
    }
    #pragma unroll
    for (int m = 16; m; m >>= 1) ssum += __shfl_xor(ssum, m, 32);
    ssum *= (1.0f / 16.0f);                        // undo 16x replication
    float lse = __logf(ssum + __expf(pos_logit));
    if (lane == 0) pair_loss[p] = lse - pos_logit;
}

// ---------------------------------------------------------------------------
// Kernel 3: deterministic single-block mean reduction.
// ---------------------------------------------------------------------------
__global__ void __launch_bounds__(256)
cpc_reduce_kernel(const float* __restrict__ pair_loss, float* __restrict__ out) {
    __shared__ float sm[256];
    float s = 0.0f;
    for (int i = threadIdx.x; i < NPAIR_; i += 256) s += pair_loss[i];
    sm[threadIdx.x] = s;
    __syncthreads();
    for (int off = 128; off; off >>= 1) {
        if ((int)threadIdx.x < off) sm[threadIdx.x] += sm[threadIdx.x + off];
        __syncthreads();
    }
    if (threadIdx.x == 0) out[0] = sm[0] * (1.0f / (float)NPAIR_);
}

// ---------------------------------------------------------------------------
extern "C" void kernel_launch(void* const* d_in, const int* in_sizes, int n_in,
                              void* d_out, int out_size, void* d_ws, size_t ws_size,
                              hipStream_t stream) {
    const float* ctx  = (const float*)d_in[0];   // (B, T, D) f32
    const float* emb  = (const float*)d_in[1];   // (B, T, D) f32
    const int*   nidx = (const int*)d_in[2];     // (Tk, B, N) i32
    float* out = (float*)d_out;

    float* ws = (float*)d_ws;
    float* inv_z = ws;                    // ROWS_ floats
    float* inv_c = ws + ROWS_;            // ROWS_ floats
    float* ploss = ws + 2 * ROWS_;        // NPAIR_ floats  (~192 KB total)

    // Kernel 1: 2*ROWS_ waves, 8 waves per 256-thread block
    int nwaves = 2 * ROWS_;
    int blocks1 = (nwaves + 7) / 8;       // 4096
    cpc_norms_kernel<<<blocks1, 256, 0, stream>>>(ctx, emb, inv_c, inv_z);

    // Kernel 2: one wave per pair, 8 waves per block -> 2024 blocks exactly
    int blocks2 = (NPAIR_ + 7) / 8;
    cpc_pair_kernel<<<blocks2, 256, 0, stream>>>(ctx, emb, nidx, inv_c, inv_z, ploss);

    // Kernel 3: mean
    cpc_reduce_kernel<<<1, 256, 0, stream>>>(ploss, out);
}